// GCN_LSTM_72138270704182
// MI455X (gfx1250) — compile-verified
//
#include <hip/hip_runtime.h>
#include <hip/hip_bf16.h>
#include <math.h>

// ---------------------------------------------------------------------------
// GCN + BiLSTM forward for MI455X (gfx1250, wave32).
// Heavy GEMMs use V_WMMA_F32_16X16X4_F32 with 32x64 register blocking:
// 8 WMMAs per K-step against 12 loaded floats/lane (~10.7 flops/byte).
// ---------------------------------------------------------------------------

typedef float v2f __attribute__((ext_vector_type(2)));
typedef float v8f __attribute__((ext_vector_type(8)));

#define Bn   16
#define Cn   64
#define Tn   256
#define Fn   75
#define Nn   (Tn * Fn)     // 19200
#define GHn  128
#define G2n  64
#define Hn   64

// ---------------------------------------------------------------------------
// Generic fp32 WMMA GEMM:  D = act( A[M,K] * B[K,N] + bias ),  row-major.
// One wave computes a 32x64 block (2 M-tiles x 4 N-tiles) of D.
// Requirements (true at every call site): M % 32 == 0, N % 64 == 0.
// TRANSB: B element [k][n] = Bm[n*ldb + k] (contiguous per-fragment loads).
// Grid: (ceil(strips/wpb), M/32, nBatch), block = 32*wpb threads.
// ---------------------------------------------------------------------------
template <int TRANSB>
__global__ __launch_bounds__(256) void wmma_gemm_f32(
    const float* __restrict__ A, const float* __restrict__ Bm,
    const float* __restrict__ bias, float* __restrict__ D,
    int M, int N, int K, int lda, int ldb, int ldd,
    long long sA, long long sB, long long sD,
    int relu)
{
  int wave = threadIdx.x >> 5;
  int lane = threadIdx.x & 31;
  int wpb  = blockDim.x >> 5;
  int n0   = (blockIdx.x * wpb + wave) << 6;       // 64-wide strip
  if (n0 >= N) return;                             // wave-uniform
  int m0   = blockIdx.y << 5;                      // 32-high block
  long long z = blockIdx.z;
  A  += z * sA;
  Bm += z * sB;
  D  += z * sD;

  int half = lane >> 4;                            // 0 or 1
  int l16  = lane & 15;

  const float* ap0 = A + (long long)(m0 + l16) * lda;
  const float* ap1 = ap0 + (long long)16 * lda;
  const float* bp0;
  const float* bp1;
  const float* bp2;
  const float* bp3;
  if (TRANSB) {
    bp0 = Bm + (long long)(n0 +  0 + l16) * ldb;
    bp1 = Bm + (long long)(n0 + 16 + l16) * ldb;
    bp2 = Bm + (long long)(n0 + 32 + l16) * ldb;
    bp3 = Bm + (long long)(n0 + 48 + l16) * ldb;
  } else {
    bp0 = Bm + n0 +  0 + l16;
    bp1 = Bm + n0 + 16 + l16;
    bp2 = Bm + n0 + 32 + l16;
    bp3 = Bm + n0 + 48 + l16;
  }

  v8f acc0[4], acc1[4];
  v8f zero8 = {0.f, 0.f, 0.f, 0.f, 0.f, 0.f, 0.f, 0.f};
#pragma unroll
  for (int j = 0; j < 4; ++j) { acc0[j] = zero8; acc1[j] = zero8; }

  int k0 = 0;
#pragma unroll 2
  for (; k0 + 4 <= K; k0 += 4) {
    int ka = k0 + half * 2;
    // streaming prefetch, one per iteration, no branches
    if (TRANSB) {
      __builtin_prefetch(bp0 + ka + 256, 0, 0);
    } else {
      __builtin_prefetch(bp0 + (long long)(ka + 16) * ldb, 0, 0);
    }
    v2f a0, a1, b0, b1, b2, b3;
    a0[0] = ap0[ka];  a0[1] = ap0[ka + 1];
    a1[0] = ap1[ka];  a1[1] = ap1[ka + 1];
    if (TRANSB) {
      b0[0] = bp0[ka]; b0[1] = bp0[ka + 1];
      b1[0] = bp1[ka]; b1[1] = bp1[ka + 1];
      b2[0] = bp2[ka]; b2[1] = bp2[ka + 1];
      b3[0] = bp3[ka]; b3[1] = bp3[ka + 1];
    } else {
      long long r0 = (long long)ka * ldb, r1 = (long long)(ka + 1) * ldb;
      b0[0] = bp0[r0]; b0[1] = bp0[r1];
      b1[0] = bp1[r0]; b1[1] = bp1[r1];
      b2[0] = bp2[r0]; b2[1] = bp2[r1];
      b3[0] = bp3[r0]; b3[1] = bp3[r1];
    }
    acc0[0] = __builtin_amdgcn_wmma_f32_16x16x4_f32(false, a0, false, b0, (short)0, acc0[0], false, false);
    acc1[0] = __builtin_amdgcn_wmma_f32_16x16x4_f32(false, a1, false, b0, (short)0, acc1[0], false, false);
    acc0[1] = __builtin_amdgcn_wmma_f32_16x16x4_f32(false, a0, false, b1, (short)0, acc0[1], false, false);
    acc1[1] = __builtin_amdgcn_wmma_f32_16x16x4_f32(false, a1, false, b1, (short)0, acc1[1], false, false);
    acc0[2] = __builtin_amdgcn_wmma_f32_16x16x4_f32(false, a0, false, b2, (short)0, acc0[2], false, false);
    acc1[2] = __builtin_amdgcn_wmma_f32_16x16x4_f32(false, a1, false, b2, (short)0, acc1[2], false, false);
    acc0[3] = __builtin_amdgcn_wmma_f32_16x16x4_f32(false, a0, false, b3, (short)0, acc0[3], false, false);
    acc1[3] = __builtin_amdgcn_wmma_f32_16x16x4_f32(false, a1, false, b3, (short)0, acc1[3], false, false);
  }
  if (k0 < K) {                                    // K tail (K = 75), zero pad
    int ka = k0 + half * 2;
    bool in0 = (ka < K), in1 = (ka + 1 < K);
    v2f a0, a1, b0, b1, b2, b3;
    a0[0] = in0 ? ap0[ka] : 0.f;  a0[1] = in1 ? ap0[ka + 1] : 0.f;
    a1[0] = in0 ? ap1[ka] : 0.f;  a1[1] = in1 ? ap1[ka + 1] : 0.f;
    if (TRANSB) {
      b0[0] = in0 ? bp0[ka] : 0.f; b0[1] = in1 ? bp0[ka + 1] : 0.f;
      b1[0] = in0 ? bp1[ka] : 0.f; b1[1] = in1 ? bp1[ka + 1] : 0.f;
      b2[0] = in0 ? bp2[ka] : 0.f; b2[1] = in1 ? bp2[ka + 1] : 0.f;
      b3[0] = in0 ? bp3[ka] : 0.f; b3[1] = in1 ? bp3[ka + 1] : 0.f;
    } else {
      long long r0 = (long long)ka * ldb, r1 = (long long)(ka + 1) * ldb;
      b0[0] = in0 ? bp0[r0] : 0.f; b0[1] = in1 ? bp0[r1] : 0.f;
      b1[0] = in0 ? bp1[r0] : 0.f; b1[1] = in1 ? bp1[r1] : 0.f;
      b2[0] = in0 ? bp2[r0] : 0.f; b2[1] = in1 ? bp2[r1] : 0.f;
      b3[0] = in0 ? bp3[r0] : 0.f; b3[1] = in1 ? bp3[r1] : 0.f;
    }
    acc0[0] = __builtin_amdgcn_wmma_f32_16x16x4_f32(false, a0, false, b0, (short)0, acc0[0], false, false);
    acc1[0] = __builtin_amdgcn_wmma_f32_16x16x4_f32(false, a1, false, b0, (short)0, acc1[0], false, false);
    acc0[1] = __builtin_amdgcn_wmma_f32_16x16x4_f32(false, a0, false, b1, (short)0, acc0[1], false, false);
    acc1[1] = __builtin_amdgcn_wmma_f32_16x16x4_f32(false, a1, false, b1, (short)0, acc1[1], false, false);
    acc0[2] = __builtin_amdgcn_wmma_f32_16x16x4_f32(false, a0, false, b2, (short)0, acc0[2], false, false);
    acc1[2] = __builtin_amdgcn_wmma_f32_16x16x4_f32(false, a1, false, b2, (short)0, acc1[2], false, false);
    acc0[3] = __builtin_amdgcn_wmma_f32_16x16x4_f32(false, a0, false, b3, (short)0, acc0[3], false, false);
    acc1[3] = __builtin_amdgcn_wmma_f32_16x16x4_f32(false, a1, false, b3, (short)0, acc1[3], false, false);
  }

  // store: C/D layout -> row = m0 + v + half*8 (+16 for second M tile),
  //        col = n0 + j*16 + l16
#pragma unroll
  for (int j = 0; j < 4; ++j) {
    int nc = n0 + j * 16 + l16;
    float bb = bias ? bias[nc] : 0.f;
#pragma unroll
    for (int v = 0; v < 8; ++v) {
      int m = m0 + v + half * 8;
      float r0 = acc0[j][v] + bb;
      float r1 = acc1[j][v] + bb;
      if (relu) { r0 = fmaxf(r0, 0.f); r1 = fmaxf(r1, 0.f); }
      D[(long long)m * ldd + nc]        = r0;
      D[(long long)(m + 16) * ldd + nc] = r1;
    }
  }
}

// ---------------------------------------------------------------------------
// Per-(b,c) mean and 1/std over N=19200 samples (for Pearson correlation).
// ---------------------------------------------------------------------------
__global__ __launch_bounds__(256) void stats_kernel(const float* __restrict__ x,
                                                    float* mu, float* rn)
{
  __shared__ float s1[256], s2[256];
  int bc = blockIdx.x;
  const float* p = x + (long long)bc * Nn;
  float a = 0.f, b = 0.f;
  for (int i = threadIdx.x; i < Nn; i += 256) { float v = p[i]; a += v; b += v * v; }
  s1[threadIdx.x] = a; s2[threadIdx.x] = b;
  __syncthreads();
  for (int o = 128; o > 0; o >>= 1) {
    if (threadIdx.x < o) { s1[threadIdx.x] += s1[threadIdx.x + o];
                           s2[threadIdx.x] += s2[threadIdx.x + o]; }
    __syncthreads();
  }
  if (threadIdx.x == 0) {
    float mean = s1[0] / (float)Nn;
    float css  = s2[0] - (float)Nn * mean * mean;       // ||x - mu||^2
    float sd   = sqrtf(fmaxf(css, 0.f) / (float)Nn);    // norm/sqrt(N)
    sd         = fmaxf(sd, 1e-6f);
    mu[bc] = mean;
    rn[bc] = 1.f / sd;
  }
}

// ---------------------------------------------------------------------------
// corr -> threshold mask -> (+eye) -> D^-1/2 A D^-1/2.
// Blocks 0..15: dynamic adjacency per batch.  Block 16: knn adjacency.
// ---------------------------------------------------------------------------
__global__ __launch_bounds__(256) void build_adj_kernel(
    const float* __restrict__ G, const float* __restrict__ mu,
    const float* __restrict__ rn, const float* __restrict__ knn,
    float* Ad, float* Ak)
{
  __shared__ float Am[Cn * Cn];
  __shared__ float dg[Cn];
  int tid = threadIdx.x, blk = blockIdx.x;
  if (blk < Bn) {
    const float* g = G + blk * Cn * Cn;
    const float* m = mu + blk * Cn;
    const float* r = rn + blk * Cn;
    for (int idx = tid; idx < Cn * Cn; idx += 256) {
      int i = idx >> 6, j = idx & 63;
      float corr = (g[idx] - (float)Nn * m[i] * m[j]) * r[i] * r[j] * (1.f / (float)Nn);
      float v = (fabsf(corr) >= 0.8f) ? 1.f : 0.f;       // threshold mask
      Am[idx] = (i == j) ? 2.f : v;                      // max(mask,eye) + eye
    }
  } else {
    for (int idx = tid; idx < Cn * Cn; idx += 256) {
      int i = idx >> 6, j = idx & 63;
      Am[idx] = knn[idx] + ((i == j) ? 1.f : 0.f);
    }
  }
  __syncthreads();
  if (tid < Cn) {
    float s = 0.f;
    for (int j = 0; j < Cn; ++j) s += Am[tid * Cn + j];
    dg[tid] = (s > 0.f) ? rsqrtf(s) : 0.f;
  }
  __syncthreads();
  float* O = (blk < Bn) ? (Ad + blk * Cn * Cn) : Ak;
  for (int idx = tid; idx < Cn * Cn; idx += 256) {
    int i = idx >> 6, j = idx & 63;
    O[idx] = dg[i] * Am[idx] * dg[j];
  }
}

// ---------------------------------------------------------------------------
// Channel attention: logits over C, softmax over C, weighted sum -> seq.
// One block per (b,t), 128 threads.
// ---------------------------------------------------------------------------
__global__ __launch_bounds__(128) void chan_attn_kernel(
    const float* __restrict__ hd, const float* __restrict__ hk,
    const float* __restrict__ wc, const float* __restrict__ bc,
    float* __restrict__ seq)
{
  __shared__ float w[Cn];
  int bt = blockIdx.x;
  int b = bt >> 8, t = bt & 255;
  int tid = threadIdx.x;
  if (tid < Cn) {
    const float* pd = hd + (((long long)(b * Cn + tid)) * Tn + t) * G2n;
    const float* pk = hk + (((long long)(b * Cn + tid)) * Tn + t) * G2n;
    float s = bc[0];
    for (int g = 0; g < G2n; ++g) s += pd[g] * wc[g] + pk[g] * wc[G2n + g];
    w[tid] = s;
  }
  __syncthreads();
  if (tid == 0) {
    float mx = w[0];
    for (int c = 1; c < Cn; ++c) mx = fmaxf(mx, w[c]);
    float sm = 0.f;
    for (int c = 0; c < Cn; ++c) { w[c] = expf(w[c] - mx); sm += w[c]; }
    float inv = 1.f / sm;
    for (int c = 0; c < Cn; ++c) w[c] *= inv;
  }
  __syncthreads();
  int g = tid & 63;
  const float* src = (tid < 64) ? hd : hk;
  float s = 0.f;
  for (int c = 0; c < Cn; ++c)
    s += w[c] * src[(((long long)(b * Cn + c)) * Tn + t) * G2n + g];
  seq[((long long)bt) * 128 + tid] = s;
}

// ---------------------------------------------------------------------------
// LSTM recurrence (gates precomputed by WMMA GEMM into xg = x*Wih^T).
// One block per (batch, direction): h,c state in LDS, 256 threads = 4H gates.
// ---------------------------------------------------------------------------
__device__ __forceinline__ float sigf(float x) { return 1.f / (1.f + expf(-x)); }

__global__ __launch_bounds__(256) void lstm_kernel(
    const float* __restrict__ xgf, const float* __restrict__ xgb,
    const float* __restrict__ whhf, const float* __restrict__ whhb,
    const float* __restrict__ bihf, const float* __restrict__ bhhf,
    const float* __restrict__ bihb, const float* __restrict__ bhhb,
    float* __restrict__ out)
{
  __shared__ float hs[Hn], cs[Hn], gs[4 * Hn];
  int b = blockIdx.x >> 1, dir = blockIdx.x & 1;
  int tid = threadIdx.x;
  const float* xg  = dir ? xgb  : xgf;
  const float* whh = dir ? whhb : whhf;
  float bias = dir ? (bihb[tid] + bhhb[tid]) : (bihf[tid] + bhhf[tid]);
  if (tid < Hn) { hs[tid] = 0.f; cs[tid] = 0.f; }
  __syncthreads();
  const float* wr = whh + tid * Hn;
  for (int stp = 0; stp < Tn; ++stp) {
    int t = dir ? (Tn - 1 - stp) : stp;
    float g = xg[((long long)(b * Tn + t)) * (4 * Hn) + tid] + bias;
    for (int k = 0; k < Hn; ++k) g += hs[k] * wr[k];
    gs[tid] = g;
    __syncthreads();
    if (tid < Hn) {
      float ig = sigf(gs[tid]);
      float fg = sigf(gs[Hn + tid]);
      float gg = tanhf(gs[2 * Hn + tid]);
      float og = sigf(gs[3 * Hn + tid]);
      float c = fg * cs[tid] + ig * gg;
      float h = og * tanhf(c);
      cs[tid] = c; hs[tid] = h;
      out[((long long)(b * Tn + t)) * (2 * Hn) + dir * Hn + tid] = h;
    }
    __syncthreads();
  }
}

// ---------------------------------------------------------------------------
// Temporal attention + classifier.  One block per batch, 256 threads (= T).
// ---------------------------------------------------------------------------
__global__ __launch_bounds__(256) void final_kernel(
    const float* __restrict__ o2, const float* __restrict__ wt,
    const float* __restrict__ bt, const float* __restrict__ Wc,
    const float* __restrict__ bcls, float* __restrict__ y)
{
  __shared__ float w[Tn];
  __shared__ float ctx[2 * Hn];
  int b = blockIdx.x, tid = threadIdx.x;
  const float* p = o2 + ((long long)(b * Tn + tid)) * (2 * Hn);
  float s = bt[0];
  for (int g = 0; g < 2 * Hn; ++g) s += p[g] * wt[g];
  w[tid] = s;
  __syncthreads();
  if (tid == 0) {
    float mx = w[0];
    for (int t = 1; t < Tn; ++t) mx = fmaxf(mx, w[t]);
    float sm = 0.f;
    for (int t = 0; t < Tn; ++t) { w[t] = expf(w[t] - mx); sm += w[t]; }
    float inv = 1.f / sm;
    for (int t = 0; t < Tn; ++t) w[t] *= inv;
  }
  __syncthreads();
  if (tid < 2 * Hn) {
    float s2 = 0.f;
    for (int t = 0; t < Tn; ++t)
      s2 += w[t] * o2[((long long)(b * Tn + t)) * (2 * Hn) + tid];
    ctx[tid] = s2;
  }
  __syncthreads();
  if (tid < 2) {
    float s3 = bcls[tid];
    for (int g = 0; g < 2 * Hn; ++g) s3 += ctx[g] * Wc[g * 2 + tid];
    y[b * 2 + tid] = s3;
  }
}

// ---------------------------------------------------------------------------
// Host launcher
// ---------------------------------------------------------------------------
static inline void launch_gemm(const float* A, const float* Bm, const float* bias,
                               float* D, int M, int N, int K,
                               int lda, int ldb, int ldd,
                               long long sA, long long sB, long long sD,
                               int nb, int transB, int relu, hipStream_t s)
{
  int strips = N >> 6;                       // N % 64 == 0 at all call sites
  int wpb    = strips < 8 ? strips : 8;
  dim3 grid((strips + wpb - 1) / wpb, M >> 5, nb);   // M % 32 == 0
  dim3 blk(32 * wpb);
  if (transB)
    wmma_gemm_f32<1><<<grid, blk, 0, s>>>(A, Bm, bias, D, M, N, K,
                                          lda, ldb, ldd, sA, sB, sD, relu);
  else
    wmma_gemm_f32<0><<<grid, blk, 0, s>>>(A, Bm, bias, D, M, N, K,
                                          lda, ldb, ldd, sA, sB, sD, relu);
}

extern "C" void kernel_launch(void* const* d_in, const int* in_sizes, int n_in,
                              void* d_out, int out_size, void* d_ws, size_t ws_size,
                              hipStream_t stream)
{
  (void)in_sizes; (void)n_in; (void)out_size; (void)ws_size;

  const float* x    = (const float*)d_in[0];
  const float* knn  = (const float*)d_in[1];
  const float* Wd1  = (const float*)d_in[2];  const float* bd1 = (const float*)d_in[3];
  const float* Wd2  = (const float*)d_in[4];  const float* bd2 = (const float*)d_in[5];
  const float* Wk1  = (const float*)d_in[6];  const float* bk1 = (const float*)d_in[7];
  const float* Wk2  = (const float*)d_in[8];  const float* bk2 = (const float*)d_in[9];
  const float* wc   = (const float*)d_in[10]; const float* bcv = (const float*)d_in[11];
  const float* wt   = (const float*)d_in[12]; const float* btv = (const float*)d_in[13];
  const float* Wcls = (const float*)d_in[14]; const float* bcl = (const float*)d_in[15];
  const float *wih[2][2], *whh[2][2], *bih[2][2], *bhh[2][2];
  for (int l = 0; l < 2; ++l)
    for (int d = 0; d < 2; ++d) {
      int base = 16 + (l * 2 + d) * 4;
      wih[l][d] = (const float*)d_in[base + 0];
      whh[l][d] = (const float*)d_in[base + 1];
      bih[l][d] = (const float*)d_in[base + 2];
      bhh[l][d] = (const float*)d_in[base + 3];
    }

  // workspace layout (float offsets)
  float* ws   = (float*)d_ws;
  float* mu   = ws + 0;         // 1024
  float* rn   = ws + 1024;      // 1024
  float* G    = ws + 2048;      // 16*64*64
  float* Ad   = ws + 67584;     // 16*64*64
  float* Ak   = ws + 133120;    // 64*64
  float* seq  = ws + 137216;    // B*T*128
  float* xgf  = ws + 661504;    // B*T*256
  float* xgb  = ws + 1710080;   // B*T*256
  float* out1 = ws + 2758656;   // B*T*128
  float* out2 = ws + 3282944;   // B*T*128
  float* big0 = ws + 3807232;   // C*T*128 (per-batch scratch)
  float* big1 = ws + 5904384;   // C*T*128 (per-batch scratch)
  float* h2d  = ws + 8001536;   // B*C*T*64
  float* h2k  = ws + 24778752;  // B*C*T*64

  // 1) correlation stats + Gram GEMM (WMMA, K=19200) + adjacency build
  stats_kernel<<<Bn * Cn, 256, 0, stream>>>(x, mu, rn);
  launch_gemm(x, x, nullptr, G, Cn, Cn, Nn, Nn, Nn, Cn,
              (long long)Cn * Nn, (long long)Cn * Nn, (long long)Cn * Cn,
              Bn, /*transB=*/1, 0, stream);
  build_adj_kernel<<<Bn + 1, 256, 0, stream>>>(G, mu, rn, knn, Ad, Ak);

  // 2) two GCN branches, two layers each; per-batch loop keeps ws small
  for (int br = 0; br < 2; ++br) {
    const float* W1 = br ? Wk1 : Wd1;  const float* b1 = br ? bk1 : bd1;
    const float* W2 = br ? Wk2 : Wd2;  const float* b2 = br ? bk2 : bd2;
    float* h2 = br ? h2k : h2d;
    for (int b = 0; b < Bn; ++b) {
      const float* xb = x + (long long)b * Cn * Tn * Fn;
      const float* Ab = br ? Ak : (Ad + b * Cn * Cn);
      // linear 1: [C*T,75] x [75,128]
      launch_gemm(xb, W1, b1, big0, Cn * Tn, GHn, Fn, Fn, GHn, GHn,
                  0, 0, 0, 1, 0, 0, stream);
      // propagate 1 + relu: [64,64] x [64, T*128]
      launch_gemm(Ab, big0, nullptr, big1, Cn, Tn * GHn, Cn,
                  Cn, Tn * GHn, Tn * GHn, 0, 0, 0, 1, 0, 1, stream);
      // linear 2: [C*T,128] x [128,64]
      launch_gemm(big1, W2, b2, big0, Cn * Tn, G2n, GHn, GHn, G2n, G2n,
                  0, 0, 0, 1, 0, 0, stream);
      // propagate 2 + relu -> h2[b]
      launch_gemm(Ab, big0, nullptr, h2 + (long long)b * Cn * Tn * G2n,
                  Cn, Tn * G2n, Cn, Cn, Tn * G2n, Tn * G2n,
                  0, 0, 0, 1, 0, 1, stream);
    }
  }

  // 3) channel attention -> seq [B,T,128]
  chan_attn_kernel<<<Bn * Tn, 128, 0, stream>>>(h2d, h2k, wc, bcv, seq);

  // 4) 2-layer BiLSTM: WMMA input-gate GEMMs + LDS-state recurrence
  const float* lin = seq;
  float* louts[2] = { out1, out2 };
  for (int l = 0; l < 2; ++l) {
    launch_gemm(lin, wih[l][0], nullptr, xgf, Bn * Tn, 4 * Hn, 2 * Hn,
                2 * Hn, 2 * Hn, 4 * Hn, 0, 0, 0, 1, /*transB=*/1, 0, stream);
    launch_gemm(lin, wih[l][1], nullptr, xgb, Bn * Tn, 4 * Hn, 2 * Hn,
                2 * Hn, 2 * Hn, 4 * Hn, 0, 0, 0, 1, /*transB=*/1, 0, stream);
    lstm_kernel<<<Bn * 2, 256, 0, stream>>>(xgf, xgb, whh[l][0], whh[l][1],
                                            bih[l][0], bhh[l][0],
                                            bih[l][1], bhh[l][1], louts[l]);
    lin = louts[l];
  }

  // 5) temporal attention + classifier -> [16,2]
  final_kernel<<<Bn, 256, 0, stream>>>(out2, wt, btv, Wcls, bcl, (float*)d_out);
}